// MoECodebook_31147102830874
// MI455X (gfx1250) — compile-verified
//
#include <hip/hip_runtime.h>
#include <hip/hip_bf16.h>
#include <math.h>

// Problem constants (match reference)
#define BB 8
#define SS 4096
#define RR 512
#define EE 64
#define KK 256
#define NATOMS 16
#define XAR 3
#define NTOK (BB * SS)   // 32768

typedef __attribute__((ext_vector_type(2))) float v2f;
typedef __attribute__((ext_vector_type(8))) float v8f;

__device__ __forceinline__ float signf_(float v) {
    return (v > 0.f) ? 1.f : ((v < 0.f) ? -1.f : 0.f);
}

// ---------------------------------------------------------------------------
// Kernel 0: zero the aux-loss accumulators (sum_p[64], cnt_pos[64])
// ---------------------------------------------------------------------------
__global__ void moe_init_kernel(float* __restrict__ accum) {
    int t = threadIdx.x;
    if (t < 128) accum[t] = 0.f;
}

// ---------------------------------------------------------------------------
// Kernel 1: expert centroids (E x R), one block per expert.
//   binary   = sign(atoms[e])                         (E, ATOMS, R)
//   idx      = argmax(combo_logits, axis=-1)          (E, K, XAR)  (first-max tie)
//   codeword = sign(sum_x w[e,k,x] * binary[e,idx])   (E, K, R)
//   centroid = mean_k codeword                        (E, R)
// ---------------------------------------------------------------------------
__global__ __launch_bounds__(256) void moe_centroid_kernel(
        const float* __restrict__ atoms,        // (E, ATOMS, R)
        const float* __restrict__ combo_w,      // (E, K, XAR)
        const float* __restrict__ combo_logits, // (E, K, XAR, ATOMS)
        float* __restrict__ cent)               // (E, R)
{
    const int e = blockIdx.x;
    const int tid = threadIdx.x;

    __shared__ int   sIdx[KK * XAR];
    __shared__ float sW[KK * XAR];

    // Phase A: argmax over ATOMS for every (k, x); load combo weight.
    for (int p = tid; p < KK * XAR; p += 256) {
        const float* lg = combo_logits + ((size_t)(e * KK) * XAR + p) * NATOMS;
        int best = 0;
        float bv = lg[0];
        #pragma unroll
        for (int a = 1; a < NATOMS; ++a) {
            float v = lg[a];
            if (v > bv) { bv = v; best = a; }   // strict > => first-max tie (argmax)
        }
        sIdx[p] = best;
        sW[p]   = combo_w[(size_t)(e * KK) * XAR + p];
    }
    __syncthreads();

    // Phase B: each thread owns r values; pack atom signs into bitmasks.
    for (int r = tid; r < RR; r += 256) {
        unsigned sb = 0u, zb = 0u;
        #pragma unroll
        for (int a = 0; a < NATOMS; ++a) {
            float v = atoms[((size_t)e * NATOMS + a) * RR + r];
            if (v < 0.f)  sb |= (1u << a);
            if (v == 0.f) zb |= (1u << a);
        }
        float acc = 0.f;
        for (int k = 0; k < KK; ++k) {
            float s = 0.f;
            #pragma unroll
            for (int x = 0; x < XAR; ++x) {
                int   i = sIdx[k * XAR + x];
                float w = sW[k * XAR + x];
                float sv = ((zb >> i) & 1u) ? 0.f : (((sb >> i) & 1u) ? -1.f : 1.f);
                s += w * sv;
            }
            acc += signf_(s);
        }
        cent[(size_t)e * RR + r] = acc * (1.f / (float)KK);
    }
}

// ---------------------------------------------------------------------------
// Kernel 2: fused router GEMM (fp32 WMMA) + softmax + top-2 + combine.
// Block = 128 threads = 4 waves; each wave owns 16 tokens x 64 experts.
// Grid = NTOK / 64 = 512 blocks.
// ---------------------------------------------------------------------------
__global__ __launch_bounds__(128) void moe_router_kernel(
        const float* __restrict__ X,     // (NTOK, R)
        const float* __restrict__ W,     // (E, R)  router weights
        const float* __restrict__ cent,  // (E, R)
        float* __restrict__ out,         // (NTOK, R)
        float* __restrict__ g_sum,       // (E) global prob sums
        float* __restrict__ g_cnt)       // (E) global positive counts
{
    __shared__ float llds[4][16][EE];    // per-wave logit tiles (16 KB)
    __shared__ int   es0[4][16], es1[4][16];
    __shared__ float gs0[4][16], gs1[4][16];
    __shared__ float bsum[EE], bcnt[EE];

    const int tid  = threadIdx.x;
    const int wave = tid >> 5;
    const int lane = tid & 31;

    if (tid < EE) { bsum[tid] = 0.f; bcnt[tid] = 0.f; }

    const int tok_base = blockIdx.x * 64 + wave * 16;
    const int m  = lane & 15;   // token row (A) / expert col (B)
    const int kh = lane >> 4;   // selects K pair {0,1} vs {2,3}

    // --- WMMA phase: logits[16 tok][64 exp], K = 512 in chunks of 4 ---
    v8f acc[4] = {};
    const float* xrow = X + (size_t)(tok_base + m) * RR + kh * 2;
    #pragma unroll 4
    for (int kc = 0; kc < RR / 4; ++kc) {
        v2f a = *(const v2f*)(xrow + kc * 4);
        #pragma unroll
        for (int t = 0; t < 4; ++t) {
            const float* wrow = W + (size_t)(t * 16 + m) * RR + kc * 4 + kh * 2;
            v2f b = *(const v2f*)wrow;
            acc[t] = __builtin_amdgcn_wmma_f32_16x16x4_f32(
                false, a, false, b, (short)0, acc[t], false, false);
        }
    }

    // Spill logit tile to LDS. C layout: VGPR j holds M=j (lanes 0-15) and
    // M=8+j (lanes 16-31); N = lane&15 within each expert tile t.
    #pragma unroll
    for (int t = 0; t < 4; ++t) {
        #pragma unroll
        for (int j = 0; j < 8; ++j) {
            int tok_m = kh * 8 + j;
            int e     = t * 16 + m;
            llds[wave][tok_m][e] = acc[t][j];
        }
    }
    __syncthreads();

    // --- Softmax + top-2 + aux stats: one token per lane (lanes 0-15) ---
    if (lane < 16) {
        const float* lg = llds[wave][lane];
        float mx = lg[0];
        for (int e = 1; e < EE; ++e) mx = fmaxf(mx, lg[e]);
        float Z = 0.f;
        for (int e = 0; e < EE; ++e) Z += expf(lg[e] - mx);
        float invZ = 1.f / Z;

        float v0 = -3.4e38f; int i0 = 0;
        for (int e = 0; e < EE; ++e)
            if (lg[e] > v0) { v0 = lg[e]; i0 = e; }
        float v1 = -3.4e38f; int i1 = 0;
        for (int e = 0; e < EE; ++e)
            if (e != i0 && lg[e] > v1) { v1 = lg[e]; i1 = e; }

        float p0 = expf(v0 - mx) * invZ;
        float p1 = expf(v1 - mx) * invZ;
        float gnorm = 1.f / (p0 + p1);
        es0[wave][lane] = i0; es1[wave][lane] = i1;
        gs0[wave][lane] = p0 * gnorm;
        gs1[wave][lane] = p1 * gnorm;

        for (int e = 0; e < EE; ++e) {
            float p = expf(lg[e] - mx) * invZ;
            atomicAdd(&bsum[e], p);
            if (p > 0.f) atomicAdd(&bcnt[e], 1.f);
        }
    }
    __syncthreads();

    if (tid < EE) {
        atomicAdd(&g_sum[tid], bsum[tid]);
        atomicAdd(&g_cnt[tid], bcnt[tid]);
    }

    // --- Combine: out[tok] = g0*cent[e0] + g1*cent[e1] (float4 streams) ---
    for (int tk = 0; tk < 16; ++tk) {
        int e0 = es0[wave][tk], e1 = es1[wave][tk];
        float g0 = gs0[wave][tk], g1 = gs1[wave][tk];
        const float4* c0 = (const float4*)(cent + (size_t)e0 * RR);
        const float4* c1 = (const float4*)(cent + (size_t)e1 * RR);
        float4* o = (float4*)(out + (size_t)(tok_base + tk) * RR);
        #pragma unroll
        for (int i = 0; i < RR / 4 / 32; ++i) {
            int idx = i * 32 + lane;
            float4 a4 = c0[idx];
            float4 b4 = c1[idx];
            float4 r;
            r.x = g0 * a4.x + g1 * b4.x;
            r.y = g0 * a4.y + g1 * b4.y;
            r.z = g0 * a4.z + g1 * b4.z;
            r.w = g0 * a4.w + g1 * b4.w;
            o[idx] = r;
        }
    }
}

// ---------------------------------------------------------------------------
// Kernel 3: finalize aux loss.
// aux = E * sum_e ( mean_bs(p_e) * mean_bs(p_e > 0) )
// ---------------------------------------------------------------------------
__global__ void moe_aux_kernel(const float* __restrict__ g_sum,
                               const float* __restrict__ g_cnt,
                               float* __restrict__ aux_out) {
    if (threadIdx.x == 0 && blockIdx.x == 0) {
        const float inv = 1.f / (float)NTOK;
        float tot = 0.f;
        for (int e = 0; e < EE; ++e)
            tot += (g_sum[e] * inv) * (g_cnt[e] * inv);
        aux_out[0] = (float)EE * tot;
    }
}

// ---------------------------------------------------------------------------
extern "C" void kernel_launch(void* const* d_in, const int* in_sizes, int n_in,
                              void* d_out, int out_size, void* d_ws, size_t ws_size,
                              hipStream_t stream) {
    (void)in_sizes; (void)n_in; (void)out_size; (void)ws_size;

    const float* x_latent     = (const float*)d_in[0];  // (B,S,R)
    const float* router_w     = (const float*)d_in[1];  // (E,R)
    const float* atoms        = (const float*)d_in[2];  // (E,ATOMS,R)
    const float* combo_w      = (const float*)d_in[3];  // (E,K,XAR)
    const float* combo_logits = (const float*)d_in[4];  // (E,K,XAR,ATOMS)

    float* out = (float*)d_out;                         // combined (B,S,R) then aux scalar
    float* aux_out = out + (size_t)NTOK * RR;

    float* cent  = (float*)d_ws;                        // E*R floats
    float* accum = cent + (size_t)EE * RR;              // 128 floats: sum_p[64], cnt[64]
    float* g_sum = accum;
    float* g_cnt = accum + EE;

    moe_init_kernel<<<1, 128, 0, stream>>>(accum);
    moe_centroid_kernel<<<EE, 256, 0, stream>>>(atoms, combo_w, combo_logits, cent);
    moe_router_kernel<<<NTOK / 64, 128, 0, stream>>>(x_latent, router_w, cent, out,
                                                     g_sum, g_cnt);
    moe_aux_kernel<<<1, 64, 0, stream>>>(g_sum, g_cnt, aux_out);
}